// MyRoPE_8899172237935
// MI455X (gfx1250) — compile-verified
//
#include <hip/hip_runtime.h>
#include <stdint.h>

// RoPE on MI455X (gfx1250): pure bandwidth problem (AI ~0.4 flop/byte).
// Strategy: per-block async-gather the scattered cos/sin diagonal of the
// rotation table into LDS (CDNA5 GLOBAL_LOAD_ASYNC_TO_LDS_B32 + s_wait_asynccnt),
// then stream x with NT b128 loads/stores.

typedef float v4f __attribute__((ext_vector_type(4)));
typedef __attribute__((address_space(1))) int gint;   // global i32
typedef __attribute__((address_space(3))) int sint;   // LDS i32

#define DK          64
#define S_TILE      32          // positions per block
#define BH_PER_BLK  4           // (batch*head) slices per block
#define THREADS     256         // 8 wave32 per block

#if defined(__AMDGCN__) && __has_builtin(__builtin_amdgcn_global_load_async_to_lds_b32)
#define ROPE_ASYNC_LDS 1
#else
#define ROPE_ASYNC_LDS 0
#endif

__device__ __forceinline__ void rope_wait_async_zero() {
#if defined(__AMDGCN__)
#if __has_builtin(__builtin_amdgcn_s_wait_asynccnt)
    __builtin_amdgcn_s_wait_asynccnt(0);
#else
    asm volatile("s_wait_asynccnt 0" ::: "memory");
#endif
#endif
}

__global__ __launch_bounds__(THREADS)
void rope_mi455x_kernel(const float* __restrict__ x,
                        const int*   __restrict__ pos_ids,
                        const float* __restrict__ table,
                        float*       __restrict__ out,
                        int S, int BH)
{
    // cs[pl*64 + 2k + 0] = cos, cs[pl*64 + 2k + 1] = sin  (8 KB)
    __shared__ float cs[S_TILE * DK];

    const int tid = threadIdx.x;
    const int s0  = blockIdx.x * S_TILE;
    const int bh0 = blockIdx.y * BH_PER_BLK;

    // ---- Phase 1: gather scattered c/s diagonal values into LDS ----------
    // S_TILE*DK = 2048 scalar values, 8 per thread. Each value lives on its
    // own cache line in the table (stride 260B), so these are true gathers;
    // the touched lines (~16MB total) stay L2-resident across the 16x reuse
    // by different blockIdx.y groups.
#pragma unroll
    for (int i = 0; i < (S_TILE * DK) / THREADS; ++i) {
        const int t    = tid + i * THREADS;
        const int pl   = t >> 6;        // 0..31 position-in-tile
        const int v    = t & 63;
        const int k    = v >> 1;        // pair index 0..31
        const int is_s = v & 1;         // 0 -> cos (row 2k), 1 -> sin (row 2k+1)
        const int s    = s0 + pl;
        if (s < S) {
            const int pos = pos_ids[s];
            // T[pos][2k + is_s][2k]
            const float* gsrc = table + ((size_t)pos * (DK * DK))
                                      + (size_t)(((k << 1) + is_s) * DK)
                                      + (size_t)(k << 1);
            const int lidx = (pl << 6) + (k << 1) + is_s;
#if ROPE_ASYNC_LDS
            __builtin_amdgcn_global_load_async_to_lds_b32(
                (gint*)gsrc, (sint*)&cs[lidx], 0, 0);
#else
            cs[lidx] = __ldg(gsrc);
#endif
        }
    }
#if ROPE_ASYNC_LDS
    rope_wait_async_zero();
#endif
    __syncthreads();

    // ---- Phase 2: stream x -> out with NT b128 accesses ------------------
    // Per (b,h) slice: S_TILE*DK floats = 512 float4 quads = 2 per thread.
#pragma unroll
    for (int j = 0; j < BH_PER_BLK; ++j) {
        const int bh = bh0 + j;
        if (bh >= BH) break;
#pragma unroll
        for (int r = 0; r < (S_TILE * DK / 4) / THREADS; ++r) {
            const int q    = tid + r * THREADS;   // 0..511
            const int pl   = q >> 4;              // position-in-tile
            const int quad = q & 15;              // float4 index within d_k
            const int s    = s0 + pl;
            if (s >= S) continue;

            const size_t off = ((size_t)bh * (size_t)S + (size_t)s) * DK
                             + (size_t)(quad << 2);

            // x/out touched once: non-temporal b128
            const v4f xv = __builtin_nontemporal_load((const v4f*)(x + off));
            const v4f cv = *(const v4f*)&cs[(pl << 6) + (quad << 2)];
            // cv = (c0, s0, c1, s1) for pairs (x.xy) and (x.zw)
            v4f o;
            o.x = fmaf(cv.x, xv.x, -cv.y * xv.y);   // c*x0 - s*x1
            o.y = fmaf(cv.y, xv.x,  cv.x * xv.y);   // s*x0 + c*x1
            o.z = fmaf(cv.z, xv.z, -cv.w * xv.w);
            o.w = fmaf(cv.w, xv.z,  cv.z * xv.w);
            __builtin_nontemporal_store(o, (v4f*)(out + off));
        }
    }
}

extern "C" void kernel_launch(void* const* d_in, const int* in_sizes, int n_in,
                              void* d_out, int out_size, void* d_ws, size_t ws_size,
                              hipStream_t stream) {
    (void)n_in; (void)out_size; (void)d_ws; (void)ws_size;
    const float* x     = (const float*)d_in[0];   // [B,H,S,64] f32
    const int*   pos   = (const int*)  d_in[1];   // [S] i32
    const float* table = (const float*)d_in[2];   // [MAX_SEQ,64,64] f32
    float*       out   = (float*)d_out;

    const int S  = in_sizes[1];
    const int BH = in_sizes[0] / (S * DK);        // batch * heads

    dim3 grid((S + S_TILE - 1) / S_TILE,
              (BH + BH_PER_BLK - 1) / BH_PER_BLK);
    rope_mi455x_kernel<<<grid, THREADS, 0, stream>>>(x, pos, table, out, S, BH);
}